// GNN_72928544686321
// MI455X (gfx1250) — compile-verified
//
#include <hip/hip_runtime.h>

#define D      128
#define D_OUT  64
#define NGRAPH 8
#define LDSP   (D + 1)   // padded row pitch for transposed W in LDS

typedef __attribute__((ext_vector_type(2))) float v2f;
typedef __attribute__((ext_vector_type(8))) float v8f;

// ---------------------------------------------------------------- utilities
__global__ __launch_bounds__(256) void k_zero(float* __restrict__ p, int n) {
  int t = blockIdx.x * 256 + threadIdx.x;
  if (t < n) p[t] = 0.0f;
}

__global__ __launch_bounds__(256) void k_degrees(const int* __restrict__ src,
                                                 const int* __restrict__ dst,
                                                 float* __restrict__ dout,
                                                 float* __restrict__ din, int nE) {
  int e = blockIdx.x * 256 + threadIdx.x;
  if (e < nE) {
    atomicAdd(dout + src[e], 1.0f);
    atomicAdd(din + dst[e], 1.0f);
  }
}

__global__ __launch_bounds__(256) void k_norms(const float* __restrict__ dout,
                                               const float* __restrict__ din,
                                               float* __restrict__ ns,
                                               float* __restrict__ nd, int M) {
  int i = blockIdx.x * 256 + threadIdx.x;
  if (i < M) {
    ns[i] = rsqrtf(fmaxf(dout[i], 1.0f));
    nd[i] = rsqrtf(fmaxf(din[i], 1.0f));
  }
}

// ------------------------------------------------- (A * diag(scale)) @ W
// A: [M,128] row-major, W: [128,128] row-major (K x N), Out: [M,128].
// Workgroup = 8 waves; block covers 32 rows x 128 cols. W is staged once into
// LDS transposed (Ws[col][k], pitch 129) so each K-step's B fragment is one
// ds_load_b64. Each wave owns two 16x16 M-tiles sharing one B fragment:
// 2 chained V_WMMA_F32_16X16X4_F32 per K-step, 64 per wave total.
__global__ __launch_bounds__(256) void k_gemm_wmma(const float* __restrict__ A,
                                                   const float* __restrict__ scale,
                                                   const float* __restrict__ W,
                                                   float* __restrict__ Out, int M) {
  __shared__ float Ws[D * LDSP];

  // cooperative transpose-stage of W (64KB) into LDS
  for (int i = threadIdx.x; i < (D * D) / 4; i += 256) {
    int idx = i << 2;
    int r = idx >> 7, c = idx & (D - 1);
    float4 w = *reinterpret_cast<const float4*>(W + idx);
    Ws[(c + 0) * LDSP + r] = w.x;
    Ws[(c + 1) * LDSP + r] = w.y;
    Ws[(c + 2) * LDSP + r] = w.z;
    Ws[(c + 3) * LDSP + r] = w.w;
  }
  __syncthreads();

  const int wave = threadIdx.x >> 5;   // 0..7 -> N tile
  const int lane = threadIdx.x & 31;
  const int lh   = lane >> 4;          // lane half (0/1)
  const int lm   = lane & 15;
  const int row0 = blockIdx.x * 32;    // two 16-row tiles per block
  const int col0 = wave * 16;

  // clamped row indices keep loads safe on the tail block without any lane
  // divergence (EXEC must stay all-ones across WMMAs)
  const int ar0 = min(row0 + lm, M - 1);
  const int ar1 = min(row0 + 16 + lm, M - 1);
  const float s0 = scale[ar0];
  const float s1 = scale[ar1];
  const float* A0 = A + (size_t)ar0 * D;
  const float* A1 = A + (size_t)ar1 * D;
  const float* Bcol = &Ws[(col0 + lm) * LDSP];

  v8f c0 = {}, c1 = {};
#pragma unroll 8
  for (int k = 0; k < D; k += 4) {
    const int ka = k + lh * 2;  // lanes 0-15: K=k,k+1 ; lanes 16-31: K=k+2,k+3
    v2f b = *reinterpret_cast<const v2f*>(Bcol + ka);  // ds_load_b64
    v2f a0, a1;
    a0.x = A0[ka] * s0;  a0.y = A0[ka + 1] * s0;
    a1.x = A1[ka] * s1;  a1.y = A1[ka + 1] * s1;
    c0 = __builtin_amdgcn_wmma_f32_16x16x4_f32(false, a0, false, b, (short)0,
                                               c0, false, false);
    c1 = __builtin_amdgcn_wmma_f32_16x16x4_f32(false, a1, false, b, (short)0,
                                               c1, false, false);
  }

  // uniform per-tile guards -> straight-line stores (M=50000: tile0 always in
  // range, tile1 skipped only by the final block)
  if (row0 + 15 < M) {
#pragma unroll
    for (int v = 0; v < 8; ++v)
      Out[(size_t)(row0 + v + lh * 8) * D + col0 + lm] = c0[v];
  }
  if (row0 + 31 < M) {
#pragma unroll
    for (int v = 0; v < 8; ++v)
      Out[(size_t)(row0 + 16 + v + lh * 8) * D + col0 + lm] = c1[v];
  }
}

// ---------------------------------------------- edge scatter: Agg[dst]+=H[src]
// 32 lanes per edge, 4 floats per lane: one 512B coalesced gather per edge,
// no-return global f32 atomics (STOREcnt path). Whole H/Agg set is L2-resident.
__global__ __launch_bounds__(256) void k_spmm(const int* __restrict__ src,
                                              const int* __restrict__ dst,
                                              const float* __restrict__ H,
                                              float* __restrict__ Agg, int nE) {
  int t = blockIdx.x * 256 + threadIdx.x;
  int e = t >> 5;
  if (e >= nE) return;
  int f  = (t & 31) << 2;
  int sN = src[e], dN = dst[e];
  const float4 v = *reinterpret_cast<const float4*>(H + (size_t)sN * D + f);
  float* o = Agg + (size_t)dN * D + f;
  atomicAdd(o + 0, v.x);
  atomicAdd(o + 1, v.y);
  atomicAdd(o + 2, v.z);
  atomicAdd(o + 3, v.w);
}

// ------------------------------------- Out = act(Agg * norm_dst[:,None] + b)
__global__ __launch_bounds__(256) void k_post(const float* __restrict__ Agg,
                                              const float* __restrict__ nd,
                                              const float* __restrict__ bias,
                                              float* __restrict__ Out, int M,
                                              int relu) {
  int t = blockIdx.x * 256 + threadIdx.x;  // one float4 group per thread
  if (t >= M * (D / 4)) return;
  int i = t >> 5;
  int f = (t & 31) << 2;
  float s  = nd[i];
  float4 v = *reinterpret_cast<const float4*>(Agg + (size_t)i * D + f);
  float4 b = *reinterpret_cast<const float4*>(bias + f);
  float4 r;
  r.x = v.x * s + b.x;
  r.y = v.y * s + b.y;
  r.z = v.z * s + b.z;
  r.w = v.w * s + b.w;
  if (relu) {
    r.x = fmaxf(r.x, 0.0f); r.y = fmaxf(r.y, 0.0f);
    r.z = fmaxf(r.z, 0.0f); r.w = fmaxf(r.w, 0.0f);
  }
  *reinterpret_cast<float4*>(Out + (size_t)i * D + f) = r;
}

// ---------------- per-graph sum of (Agg*norm_dst + b2) staged through LDS
__global__ __launch_bounds__(256) void k_pool(const float* __restrict__ Agg,
                                              const float* __restrict__ nd,
                                              const float* __restrict__ b2,
                                              const int* __restrict__ gid,
                                              float* __restrict__ pooled,  // [8,128]
                                              float* __restrict__ cnt,     // [8]
                                              int M) {
  __shared__ float lpool[NGRAPH * D];
  __shared__ float lcnt[NGRAPH];
  for (int i = threadIdx.x; i < NGRAPH * D; i += 256) lpool[i] = 0.0f;
  if (threadIdx.x < NGRAPH) lcnt[threadIdx.x] = 0.0f;
  __syncthreads();

  const int NODES_PER_BLOCK = 64;
  int base = blockIdx.x * NODES_PER_BLOCK;
  int f    = threadIdx.x & (D - 1);  // feature
  int sub  = threadIdx.x >> 7;       // 0..1 (two nodes in flight)
  for (int n = sub; n < NODES_PER_BLOCK; n += 2) {
    int i = base + n;
    if (i >= M) break;
    int   g = gid[i];
    float v = Agg[(size_t)i * D + f] * nd[i] + b2[f];
    atomicAdd(&lpool[g * D + f], v);               // ds_add_f32
    if (f == 0) atomicAdd(&lcnt[g], 1.0f);
  }
  __syncthreads();

  for (int i = threadIdx.x; i < NGRAPH * D; i += 256) {
    float v = lpool[i];
    if (v != 0.0f) atomicAdd(&pooled[i], v);       // graph_id sorted -> few hits
  }
  if (threadIdx.x < NGRAPH && lcnt[threadIdx.x] != 0.0f)
    atomicAdd(&cnt[threadIdx.x], lcnt[threadIdx.x]);
}

// ---------------------------- out = (pooled/cnt) @ Wl + bl   ([8,128]@[128,64])
__global__ __launch_bounds__(256) void k_head(const float* __restrict__ pooled,
                                              const float* __restrict__ cnt,
                                              const float* __restrict__ Wl,
                                              const float* __restrict__ bl,
                                              float* __restrict__ out) {
  int t = blockIdx.x * 256 + threadIdx.x;
  if (t >= NGRAPH * D_OUT) return;
  int g = t / D_OUT, o = t % D_OUT;
  float inv = 1.0f / fmaxf(cnt[g], 1.0f);
  float acc = bl[o];
#pragma unroll 8
  for (int k = 0; k < D; ++k)
    acc += (pooled[g * D + k] * inv) * Wl[(size_t)k * D_OUT + o];
  out[t] = acc;
}

// ---------------------------------------------------------------- dispatcher
extern "C" void kernel_launch(void* const* d_in, const int* in_sizes, int n_in,
                              void* d_out, int out_size, void* d_ws, size_t ws_size,
                              hipStream_t stream) {
  const float* x  = (const float*)d_in[0];
  const float* W1 = (const float*)d_in[1];
  const float* b1 = (const float*)d_in[2];
  const float* W2 = (const float*)d_in[3];
  const float* b2 = (const float*)d_in[4];
  const float* Wl = (const float*)d_in[5];
  const float* bl = (const float*)d_in[6];
  const int* src  = (const int*)d_in[7];
  const int* dst  = (const int*)d_in[8];
  const int* gid  = (const int*)d_in[9];

  const int M  = in_sizes[0] / D;  // 50000
  const int nE = in_sizes[7];      // 800000

  // Workspace layout (float offsets, small region zeroed every call):
  float* ws       = (float*)d_ws;
  float* deg_out  = ws;                 // [<=51200]
  float* deg_in   = ws + 51200;
  float* norm_src = ws + 102400;
  float* norm_dst = ws + 153600;
  float* pooled   = ws + 204800;        // [8*128]
  float* cnt      = ws + 205824;        // [8]
  float* bufA     = ws + 206848;        // [M*128]
  float* bufB     = bufA + (size_t)M * D;

  const int FRONT = 206848;
  const int NF    = M * D;
  const int GEMM_GRID = (M + 31) / 32;

  // zero degree/pool accumulators + bufB (layer-1 scatter destination)
  k_zero<<<(FRONT + 255) / 256, 256, 0, stream>>>(ws, FRONT);
  k_zero<<<(NF + 255) / 256, 256, 0, stream>>>(bufB, NF);

  // degrees -> norms
  k_degrees<<<(nE + 255) / 256, 256, 0, stream>>>(src, dst, deg_out, deg_in, nE);
  k_norms<<<(M + 255) / 256, 256, 0, stream>>>(deg_out, deg_in, norm_src, norm_dst, M);

  // layer 1: h = (x*norm_src)@W1 ; agg = scatter ; h1 = relu(agg*norm_dst+b1)
  k_gemm_wmma<<<GEMM_GRID, 256, 0, stream>>>(x, norm_src, W1, bufA, M);
  k_spmm<<<(nE * 32 + 255) / 256, 256, 0, stream>>>(src, dst, bufA, bufB, nE);
  k_post<<<(M * 32 + 255) / 256, 256, 0, stream>>>(bufB, norm_dst, b1, bufA, M, 1);

  // layer 2: h2 = (h1*norm_src)@W2 into bufB (agg1 consumed), re-zero bufA,
  // scatter into bufA
  k_gemm_wmma<<<GEMM_GRID, 256, 0, stream>>>(bufA, norm_src, W2, bufB, M);
  k_zero<<<(NF + 255) / 256, 256, 0, stream>>>(bufA, NF);
  k_spmm<<<(nE * 32 + 255) / 256, 256, 0, stream>>>(src, dst, bufB, bufA, nE);

  // pooling (applies *norm_dst + b2 per node) and linear head
  k_pool<<<(M + 63) / 64, 256, 0, stream>>>(bufA, norm_dst, b2, gid, pooled, cnt, M);
  k_head<<<(NGRAPH * D_OUT + 255) / 256, 256, 0, stream>>>(pooled, cnt, Wl, bl,
                                                           (float*)d_out);
}